// BahdanauAttention_53558242181734
// MI455X (gfx1250) — compile-verified
//
#include <hip/hip_runtime.h>
#include <math.h>

// Problem constants (match reference)
#define B_    8
#define T_    128
#define S_    1024
#define ENC_  512
#define DEC_  512
#define ATT_  256

typedef __attribute__((ext_vector_type(2))) float v2f;
typedef __attribute__((ext_vector_type(8))) float v8f;

// ---------------------------------------------------------------------------
// Fast tanh: prefer the gfx1250 hardware V_TANH_F32 transcendental; otherwise
// a 3-op exp/rcp formulation (exact at saturation: +/-1), never OCML's long
// polynomial expansion.  This sits in the dominant 268M-element loop.
// ---------------------------------------------------------------------------
__device__ __forceinline__ float fast_tanh(float x) {
#if __has_builtin(__builtin_amdgcn_tanhf)
  return __builtin_amdgcn_tanhf(x);              // v_tanh_f32 (TRANS pipe)
#else
  const float e = __expf(2.0f * x);              // v_exp_f32
  return 1.0f - 2.0f * __builtin_amdgcn_rcpf(e + 1.0f);  // v_rcp_f32
#endif
}

// ---------------------------------------------------------------------------
// One-wave 16x16 fp32 GEMM tile using V_WMMA_F32_16X16X4_F32.
// C(16x16) += A(16xK) * B(Kx16), arbitrary strides (elements).
// ISA fragment layouts (CDNA5 ISA):
//   A 16x4 : lanes 0-15 -> M=lane, VGPR0=K0,VGPR1=K1 ; lanes 16-31 -> K2,K3
//   B 4x16 : lanes 0-15 -> N=lane, K=0..1           ; lanes 16-31 -> K=2..3
//   C/D    : vgpr r -> M=r (lanes 0-15) / M=r+8 (lanes 16-31), N=lane%16
// ---------------------------------------------------------------------------
__device__ __forceinline__ v8f wmma16x16_f32(const float* __restrict__ A,
                                             int a_sM, int a_sK,
                                             const float* __restrict__ Bm,
                                             int b_sK, int b_sN,
                                             int K, v8f c) {
  const int lane = threadIdx.x & 31;
  const int l16  = lane & 15;
  const int kh   = (lane >> 4) << 1;  // 0 for lanes 0-15, 2 for lanes 16-31
  for (int k0 = 0; k0 < K; k0 += 4) {
    const int k = k0 + kh;
    v2f a, b;
    a.x = A[l16 * a_sM + (k    ) * a_sK];
    a.y = A[l16 * a_sM + (k + 1) * a_sK];
    b.x = Bm[(k    ) * b_sK + l16 * b_sN];
    b.y = Bm[(k + 1) * b_sK + l16 * b_sN];
    // D = A*B + C  (fp32, 16x16x4)
    c = __builtin_amdgcn_wmma_f32_16x16x4_f32(false, a, false, b,
                                              (short)0, c, false, false);
  }
  return c;
}

// ---------------------------------------------------------------------------
// Stage 1: kp_t[b][a][s] = sum_e keys[b][s][e]*Kw[a][e] + Kb[a]
// One wave per 16x16 (a,s) tile.  grid = (S/16, ATT/16, B), block = 32.
// ---------------------------------------------------------------------------
__global__ void kp_gemm_kernel(const float* __restrict__ keys,
                               const float* __restrict__ Kw,
                               const float* __restrict__ Kb,
                               float* __restrict__ kp_t) {
  const int s0 = blockIdx.x * 16;
  const int a0 = blockIdx.y * 16;
  const int b  = blockIdx.z;
  const float* A  = Kw + a0 * ENC_;                 // A[m][k] = Kw[a0+m][e]
  const float* Bm = keys + (b * S_ + s0) * ENC_;    // B[k][n] = keys[b][s0+n][e]
  v8f c = {};
  c = wmma16x16_f32(A, ENC_, 1, Bm, 1, ENC_, ENC_, c);
  const int lane = threadIdx.x & 31;
  const int l16  = lane & 15;
  const int mh   = (lane >> 4) * 8;
#pragma unroll
  for (int r = 0; r < 8; ++r) {
    const int a = a0 + r + mh;
    kp_t[(b * ATT_ + a) * S_ + s0 + l16] = c[r] + Kb[a];
  }
}

// ---------------------------------------------------------------------------
// Stage 2: u_t[b][a][t] = sum_d queries[b][t][d]*Uw[a][d] + Ub[a]
// grid = (T/16, ATT/16, B), block = 32.
// ---------------------------------------------------------------------------
__global__ void u_gemm_kernel(const float* __restrict__ queries,
                              const float* __restrict__ Uw,
                              const float* __restrict__ Ub,
                              float* __restrict__ u_t) {
  const int t0 = blockIdx.x * 16;
  const int a0 = blockIdx.y * 16;
  const int b  = blockIdx.z;
  const float* A  = Uw + a0 * DEC_;                  // A[m][k] = Uw[a0+m][d]
  const float* Bm = queries + (b * T_ + t0) * DEC_;  // B[k][n] = queries[b][t0+n][d]
  v8f c = {};
  c = wmma16x16_f32(A, DEC_, 1, Bm, 1, DEC_, DEC_, c);
  const int lane = threadIdx.x & 31;
  const int l16  = lane & 15;
  const int mh   = (lane >> 4) * 8;
#pragma unroll
  for (int r = 0; r < 8; ++r) {
    const int a = a0 + r + mh;
    u_t[(b * ATT_ + a) * T_ + t0 + l16] = c[r] + Ub[a];
  }
}

// ---------------------------------------------------------------------------
// Stage 3 (fused, dominant): per (b,t):
//   score[s] = vb + sum_a vw[a]*tanh(kp_t[b,a,s] + u[b,t,a]);  mask; softmax.
// Never materializes [B,T,S,ATT].  grid = (T, B), block = 256 (8 waves).
// Each thread owns 4 s-values; kp_t reads are lane-contiguous in s.
// Hot loop per element: ds_load_b64 (vw,u pair) + global_load_b32 (kp)
//                       + v_add + v_tanh (TRANS) + v_fmac.
// ---------------------------------------------------------------------------
__global__ void score_softmax_kernel(const float* __restrict__ kp_t,
                                     const float* __restrict__ u_t,
                                     const float* __restrict__ vw,
                                     const float* __restrict__ vb,
                                     const int* __restrict__ mask,
                                     float* __restrict__ alpha) {
  const int t   = blockIdx.x;
  const int b   = blockIdx.y;
  const int tid = threadIdx.x;
  __shared__ float2 wu_sh[ATT_];   // .x = vw[a], .y = u[b][t][a]
  __shared__ float  red[256];

  wu_sh[tid] = make_float2(vw[tid], u_t[(b * ATT_ + tid) * T_ + t]);
  __syncthreads();

  const float vb0 = vb[0];
  float sc[4];
#pragma unroll
  for (int i = 0; i < 4; ++i) {
    const int s = tid + 256 * i;
    const float* kpp = kp_t + (size_t)b * ATT_ * S_ + s;  // element a: kpp[a*S_]
    float acc = 0.f;
#pragma unroll 4
    for (int a = 0; a < ATT_; ++a) {
      const float2 wu = wu_sh[a];
      acc = __builtin_fmaf(wu.x, fast_tanh(kpp[a * S_] + wu.y), acc);
    }
    sc[i] = acc + vb0;
    if (mask[b * S_ + s] == 0) sc[i] = -1.0e9f;
  }

  // block max-reduce over 1024 scores
  float m = fmaxf(fmaxf(sc[0], sc[1]), fmaxf(sc[2], sc[3]));
  red[tid] = m;
  __syncthreads();
  for (int off = 128; off > 0; off >>= 1) {
    if (tid < off) red[tid] = fmaxf(red[tid], red[tid + off]);
    __syncthreads();
  }
  const float gmax = red[0];
  __syncthreads();

  float e[4];
  float psum = 0.f;
#pragma unroll
  for (int i = 0; i < 4; ++i) {
    e[i] = __expf(sc[i] - gmax);
    psum += e[i];
  }
  red[tid] = psum;
  __syncthreads();
  for (int off = 128; off > 0; off >>= 1) {
    if (tid < off) red[tid] += red[tid + off];
    __syncthreads();
  }
  const float inv = 1.0f / red[0];

  float* ap = alpha + ((size_t)b * T_ + t) * S_;
#pragma unroll
  for (int i = 0; i < 4; ++i) ap[tid + 256 * i] = e[i] * inv;
}

// ---------------------------------------------------------------------------
// Stage 4: context[b][t][v] = sum_s alpha[b][t][s]*values[b][s][v]
// grid = (ENC/16, T/16, B), block = 32.  B-fragment loads are coalesced in v.
// ---------------------------------------------------------------------------
__global__ void ctx_gemm_kernel(const float* __restrict__ alpha,
                                const float* __restrict__ values,
                                float* __restrict__ out) {
  const int v0 = blockIdx.x * 16;
  const int t0 = blockIdx.y * 16;
  const int b  = blockIdx.z;
  const float* A  = alpha + ((size_t)b * T_ + t0) * S_;  // A[m][k] = alpha[b][t0+m][s]
  const float* Bm = values + (size_t)b * S_ * ENC_ + v0; // B[k][n] = values[b][s][v0+n]
  v8f c = {};
  c = wmma16x16_f32(A, S_, 1, Bm, ENC_, 1, S_, c);
  const int lane = threadIdx.x & 31;
  const int l16  = lane & 15;
  const int mh   = (lane >> 4) * 8;
#pragma unroll
  for (int r = 0; r < 8; ++r) {
    out[((size_t)b * T_ + t0 + r + mh) * ENC_ + v0 + l16] = c[r];
  }
}

// ---------------------------------------------------------------------------
extern "C" void kernel_launch(void* const* d_in, const int* in_sizes, int n_in,
                              void* d_out, int out_size, void* d_ws, size_t ws_size,
                              hipStream_t stream) {
  const float* queries = (const float*)d_in[0];  // [B,T,DEC]
  const float* keys    = (const float*)d_in[1];  // [B,S,ENC]
  const float* values  = (const float*)d_in[2];  // [B,S,ENC]
  const int*   mask    = (const int*)  d_in[3];  // [B,S]
  const float* Kw      = (const float*)d_in[4];  // [ATT,ENC]
  const float* Kb      = (const float*)d_in[5];  // [ATT]
  const float* Uw      = (const float*)d_in[6];  // [ATT,DEC]
  const float* Ub      = (const float*)d_in[7];  // [ATT]
  const float* vw      = (const float*)d_in[8];  // [ATT]
  const float* vb      = (const float*)d_in[9];  // scalar
  float* out = (float*)d_out;                    // [B,T,ENC]

  // Workspace: kp_t (8 MB) | u_t (1 MB) | alpha (4 MB)
  char* ws = (char*)d_ws;
  float* kp_t  = (float*)(ws);
  float* u_t   = (float*)(ws + (size_t)B_ * ATT_ * S_ * sizeof(float));
  float* alpha = (float*)(ws + (size_t)B_ * ATT_ * S_ * sizeof(float)
                             + (size_t)B_ * ATT_ * T_ * sizeof(float));

  kp_gemm_kernel<<<dim3(S_ / 16, ATT_ / 16, B_), 32, 0, stream>>>(keys, Kw, Kb, kp_t);
  u_gemm_kernel<<<dim3(T_ / 16, ATT_ / 16, B_), 32, 0, stream>>>(queries, Uw, Ub, u_t);
  score_softmax_kernel<<<dim3(T_, B_), 256, 0, stream>>>(kp_t, u_t, vw, vb, mask, alpha);
  ctx_gemm_kernel<<<dim3(ENC_ / 16, T_ / 16, B_), 32, 0, stream>>>(alpha, values, out);
}